// Block_38517266710836
// MI455X (gfx1250) — compile-verified
//
#include <hip/hip_runtime.h>
#include <hip/hip_bf16.h>
#include <math.h>
#include <stdint.h>

// ---------------------------------------------------------------------------
// Problem constants (from the reference)
// ---------------------------------------------------------------------------
#define BATCH   4
#define CPD     64
#define TIME    65536
#define WINDOWN 2048
#define STEPN   1024
#define NCOEF   1025          // WINDOW/2 + 1
#define NFFT    1024          // complex FFT length (packed real FFT)
#define FRAMES  64            // (TIME + STEP - WINDOW)/STEP + 1

#define TWO_PI 6.28318530717958647692f

typedef __attribute__((ext_vector_type(2))) float v2f;
typedef __attribute__((ext_vector_type(8))) float v8f;
typedef __attribute__((ext_vector_type(4))) int   v4i;

#define AS1 __attribute__((address_space(1)))
#define AS3 __attribute__((address_space(3)))

// gfx1250 async global->LDS DMA path (ASYNCcnt), guarded so the file compiles
// on toolchains where the builtin is absent (falls back to plain loads).
// Probe-confirmed signature: param0 = int4 AS1* (non-const), param1 = LDS ptr.
#if defined(__AMDGCN__) && \
    __has_builtin(__builtin_amdgcn_global_load_async_to_lds_b128) && \
    __has_builtin(__builtin_amdgcn_s_wait_asynccnt)
#define HAVE_ASYNC_LDS 1
#else
#define HAVE_ASYNC_LDS 0
#endif

// ---------------------------------------------------------------------------
// Kernel 1: channel mix  out[b,d,t] = sum_c x[b,c,t] * M[c,d]
// FP32 WMMA (V_WMMA_F32_16X16X4_F32). One wave owns a 16-wide d tile and
// 8 consecutive 16-wide t tiles; the A operand (M^T tile, 16 v2f regs) is
// loaded ONCE per wave and reused for all 8 tiles (x fits in the 192MB L2,
// so the 4x d-tile re-read of x is L2-resident). Exact grid -> EXEC all ones.
//
// A (16x4 f32) lane layout: lanes 0-15 row=lane, VGPR0={K=0|K=2}, VGPR1={K=1|K=3}
// B (4x16 f32):            VGPR0 = {K=0 lanes0-15 | K=2 lanes16-31}, VGPR1 = {K=1|K=3}
// C/D (16x16 f32): VGPR v, lane l -> row = v + 8*(l>>4), col = l&15
// ---------------------------------------------------------------------------
#define TT 8   // t-tiles per wave

__global__ __launch_bounds__(256) void mix_wmma(const float* __restrict__ x,
                                                const float* __restrict__ M,
                                                float* __restrict__ out) {
  const int wave = (blockIdx.x * blockDim.x + threadIdx.x) >> 5;
  const int lane = threadIdx.x & 31;

  const int group = wave & 511;                 // 4096/TT = 512 tile-groups
  const int rest  = wave >> 9;
  const int dtile = rest & 3;                   // 4 d-tiles
  const int b     = rest >> 2;                  // 4 batches

  const int tbase = group * (16 * TT);
  const int d0    = dtile * 16;
  const int row   = lane & 15;                  // A row (d) / B,C,D column (t)
  const int hi    = lane >> 4;

  // Preload A = M^T[d0+row, :] into registers (reused for all TT tiles).
  v2f a[16];
#pragma unroll
  for (int cc = 0; cc < 16; ++cc) {
    const int ka = cc * 4 + 2 * hi;
    a[cc].x = M[(ka + 0) * CPD + (d0 + row)];
    a[cc].y = M[(ka + 1) * CPD + (d0 + row)];
  }

  const float* xb = x + (size_t)b * CPD * TIME;
  float* ob       = out + (size_t)b * CPD * TIME;

  for (int tt = 0; tt < TT; ++tt) {
    const int t0 = tbase + tt * 16;
    v8f acc = {};
#pragma unroll
    for (int cc = 0; cc < 16; ++cc) {
      const int ka = cc * 4 + 2 * hi;
      v2f bb;
      bb.x = xb[(size_t)(ka + 0) * TIME + t0 + row];
      bb.y = xb[(size_t)(ka + 1) * TIME + t0 + row];
      acc = __builtin_amdgcn_wmma_f32_16x16x4_f32(
          /*neg_a=*/false, a[cc], /*neg_b=*/false, bb,
          /*c_mod=*/(short)0, acc, /*reuse_a=*/false, /*reuse_b=*/false);
    }
#pragma unroll
    for (int v = 0; v < 8; ++v)
      ob[(size_t)(d0 + v + 8 * hi) * TIME + t0 + row] = acc[v];
  }
}

// ---------------------------------------------------------------------------
// Shared radix-2 Stockham (auto-sorting) 1024-pt complex FFT core.
// 256 threads, 512 butterflies/stage, 10 stages, LDS ping-pong.
// tw[m] = exp(sign*2*pi*i*m/1024), m = 0..511 (sign baked into the table).
// ---------------------------------------------------------------------------
__device__ __forceinline__ float2* fft1024_stockham(float2* S, float2* D,
                                                    const float2* tw, int tid) {
  for (int stage = 0; stage < 10; ++stage) {
    const int Ns = 1 << stage;
    __syncthreads();
#pragma unroll
    for (int jj = 0; jj < 2; ++jj) {
      const int j = tid + jj * 256;             // j in [0, 512)
      const int r = j & (Ns - 1);
      const float2 w  = tw[r << (9 - stage)];   // r * (512/Ns)
      const float2 v0 = S[j];
      const float2 v1 = S[j + 512];
      const float2 v1w = make_float2(v1.x * w.x - v1.y * w.y,
                                     v1.x * w.y + v1.y * w.x);
      const int idxD = ((j >> stage) << (stage + 1)) + r;
      D[idxD]      = make_float2(v0.x + v1w.x, v0.y + v1w.y);
      D[idxD + Ns] = make_float2(v0.x - v1w.x, v0.y - v1w.y);
    }
    float2* t = S; S = D; D = t;
  }
  __syncthreads();
  return S;
}

// ---------------------------------------------------------------------------
// Kernel 2: window + pack + forward rfft(2048) per (b,d,frame).
// Frame input (8KB contiguous) is staged into LDS with the gfx1250 async
// global->LDS DMA (2 x B128 per lane, ASYNCcnt), then windowed/packed.
// blockIdx.x = ((b*64 + d)*64 + f).  Writes 1025 complex bins.
// ---------------------------------------------------------------------------
__global__ __launch_bounds__(256) void fwd_fft(const float* __restrict__ mixed,
                                               float2* __restrict__ spec) {
  __shared__ __align__(16) float raw[WINDOWN];
  __shared__ float2 bufA[NFFT];
  __shared__ float2 bufB[NFFT];
  __shared__ float2 tw[NFFT / 2];

  const int tid = threadIdx.x;
  const int f   = blockIdx.x & 63;
  const size_t bd = (size_t)(blockIdx.x >> 6);  // b*64 + d
  const float* src = mixed + bd * TIME + (size_t)f * STEPN;
  const int tlim = TIME - f * STEPN;            // valid samples in this frame

#if HAVE_ASYNC_LDS
  {
    // 256 lanes x 2 x 16B = 8KB = whole frame. Tail frame over-reads at most
    // 4KB past `mixed`, which stays inside the workspace; masked below.
    const int i0 = tid * 4;
    const int i1 = i0 + 1024;
    __builtin_amdgcn_global_load_async_to_lds_b128(
        (AS1 v4i*)(uintptr_t)(src + i0),
        (AS3 v4i*)(uint32_t)(uintptr_t)&raw[i0], 0, 0);
    __builtin_amdgcn_global_load_async_to_lds_b128(
        (AS1 v4i*)(uintptr_t)(src + i1),
        (AS3 v4i*)(uint32_t)(uintptr_t)&raw[i1], 0, 0);
    __builtin_amdgcn_s_wait_asynccnt(0);
  }
  __syncthreads();
#else
  for (int ii = tid; ii < WINDOWN; ii += 256)
    raw[ii] = (ii < tlim) ? src[ii] : 0.0f;
  __syncthreads();
#endif

  const float wstep = TWO_PI / (float)WINDOWN;
#pragma unroll
  for (int nn = 0; nn < 4; ++nn) {
    const int n  = tid + nn * 256;              // n in [0, 1024)
    const int i0 = 2 * n, i1 = 2 * n + 1;
    const float s0 = (i0 < tlim) ? raw[i0] : 0.0f;
    const float s1 = (i1 < tlim) ? raw[i1] : 0.0f;
    const float w0 = 0.5f - 0.5f * __cosf(wstep * (float)i0);
    const float w1 = 0.5f - 0.5f * __cosf(wstep * (float)i1);
    bufA[n] = make_float2(s0 * w0, s1 * w1);    // z[n] = xw[2n] + i*xw[2n+1]
  }
#pragma unroll
  for (int mm = 0; mm < 2; ++mm) {
    const int m = tid + mm * 256;               // m in [0, 512)
    float s, c;
    __sincosf(-TWO_PI * (float)m / (float)NFFT, &s, &c);
    tw[m] = make_float2(c, s);
  }

  float2* Z = fft1024_stockham(bufA, bufB, tw, tid);

  // Real-FFT untangle: X[k] = Fe[k] + e^{-2pi i k / 2048} * Fo[k], k=0..1024
  const size_t outBase = (size_t)blockIdx.x * NCOEF;
#pragma unroll
  for (int kk = 0; kk < 5; ++kk) {
    const int k = tid + kk * 256;               // covers 0..1279; clip at 1024
    if (k <= NFFT) {
      const float2 zk = Z[k & (NFFT - 1)];      // Z[1024] wraps to Z[0]
      const float2 zn = Z[(NFFT - k) & (NFFT - 1)];
      const float2 Fe = make_float2(0.5f * (zk.x + zn.x), 0.5f * (zk.y - zn.y));
      const float2 Fo = make_float2(0.5f * (zk.y + zn.y), 0.5f * (zn.x - zk.x));
      float s, c;
      __sincosf(-TWO_PI * (float)k / (float)WINDOWN, &s, &c);
      spec[outBase + k] = make_float2(Fe.x + c * Fo.x - s * Fo.y,
                                      Fe.y + c * Fo.y + s * Fo.x);
    }
  }
}

// ---------------------------------------------------------------------------
// Kernel 3: sequential scan over frames (in place):
//   out[f] = (spec[f] + out[f-1]) * T[d,k]   (T is real)
// One thread per (b,d,k): 4*64*1025 = 262400 independent recurrences.
// ---------------------------------------------------------------------------
__global__ __launch_bounds__(256) void scan_frames(float2* __restrict__ spec,
                                                   const float* __restrict__ transfer) {
  const int idx = blockIdx.x * blockDim.x + threadIdx.x;
  if (idx >= BATCH * CPD * NCOEF) return;
  const int k  = idx % NCOEF;
  const int bd = idx / NCOEF;
  const int d  = bd & (CPD - 1);
  const float T = transfer[d * NCOEF + k];

  float2 carry = make_float2(0.0f, 0.0f);
  const size_t base = (size_t)bd * FRAMES * NCOEF + k;
#pragma unroll 4
  for (int f = 0; f < FRAMES; ++f) {
    float2* p = spec + base + (size_t)f * NCOEF;
    if (f + 1 < FRAMES) __builtin_prefetch(p + NCOEF, 0, 0);  // global_prefetch_b8
    const float2 s = *p;
    carry.x = (s.x + carry.x) * T;
    carry.y = (s.y + carry.y) * T;
    *p = carry;
  }
}

// ---------------------------------------------------------------------------
// Kernel 4: irfft(2048) + synthesis window per (b,d,frame).
// Spectrum staged once into LDS (each bin is consumed twice: X[k], X[1024-k]).
//   Fe = (X[k]+conj(X[1024-k]))/2 ; Fo = (X[k]-conj(X[1024-k]))/2 * e^{+2pi i k/2048}
//   Z[k] = Fe + i*Fo ; z = ifft1024(Z) ; x[2n]=Re z[n], x[2n+1]=Im z[n]
// ---------------------------------------------------------------------------
__global__ __launch_bounds__(256) void inv_fft(const float2* __restrict__ spec,
                                               float* __restrict__ audio) {
  __shared__ float2 sp[NCOEF];
  __shared__ float2 bufA[NFFT];
  __shared__ float2 bufB[NFFT];
  __shared__ float2 tw[NFFT / 2];

  const int tid = threadIdx.x;
  const size_t inBase = (size_t)blockIdx.x * NCOEF;

  for (int k = tid; k < NCOEF; k += 256) sp[k] = spec[inBase + k];
#pragma unroll
  for (int mm = 0; mm < 2; ++mm) {
    const int m = tid + mm * 256;
    float s, c;
    __sincosf(TWO_PI * (float)m / (float)NFFT, &s, &c); // conjugate twiddles
    tw[m] = make_float2(c, s);
  }
  __syncthreads();

#pragma unroll
  for (int kk = 0; kk < 4; ++kk) {
    const int k = tid + kk * 256;               // k in [0, 1024)
    const float2 Xk = sp[k];
    const float2 Xn = sp[NFFT - k];             // X[1024-k], k=0 -> X[1024]
    const float2 Fe = make_float2(0.5f * (Xk.x + Xn.x), 0.5f * (Xk.y - Xn.y));
    const float2 G  = make_float2(0.5f * (Xk.x - Xn.x), 0.5f * (Xk.y + Xn.y));
    float s, c;
    __sincosf(TWO_PI * (float)k / (float)WINDOWN, &s, &c);
    const float2 Fo = make_float2(G.x * c - G.y * s, G.x * s + G.y * c);
    bufA[k] = make_float2(Fe.x - Fo.y, Fe.y + Fo.x);   // Fe + i*Fo
  }

  float2* z = fft1024_stockham(bufA, bufB, tw, tid);

  float* dst = audio + (size_t)blockIdx.x * WINDOWN;
  const float invN  = 1.0f / (float)NFFT;       // ifft normalization (1/1024)
  const float wstep = TWO_PI / (float)WINDOWN;
#pragma unroll
  for (int nn = 0; nn < 4; ++nn) {
    const int n  = tid + nn * 256;
    const float2 zz = z[n];
    const int i0 = 2 * n, i1 = 2 * n + 1;
    const float w0 = 0.5f - 0.5f * __cosf(wstep * (float)i0);
    const float w1 = 0.5f - 0.5f * __cosf(wstep * (float)i1);
    dst[i0] = zz.x * invN * w0;
    dst[i1] = zz.y * invN * w1;
  }
}

// ---------------------------------------------------------------------------
// Kernel 5: overlap-add (exactly 2 frames/sample at 50% overlap) + gain + tanh
// ---------------------------------------------------------------------------
__global__ __launch_bounds__(256) void overlap_add(const float* __restrict__ audio,
                                                   const float* __restrict__ gain,
                                                   float* __restrict__ out) {
  const size_t idx = (size_t)blockIdx.x * blockDim.x + threadIdx.x;
  if (idx >= (size_t)BATCH * CPD * TIME) return;
  const int t   = (int)(idx & (TIME - 1));
  const size_t bd = idx >> 16;                  // b*64 + d
  const int q = t >> 10;                        // frame index floor(t/1024)
  const int r = t & (STEPN - 1);

  const float* base = audio + bd * (size_t)FRAMES * WINDOWN;
  float s = base[(size_t)q * WINDOWN + r];
  if (q >= 1) s += base[(size_t)(q - 1) * WINDOWN + r + STEPN];
  out[idx] = tanhf(s * gain[0]);
}

// ---------------------------------------------------------------------------
// Launcher. Inputs: d_in[0]=x (4,64,65536) f32, d_in[1]=transfer (64,1025) f32,
// d_in[2]=mixer_matrix (64,64) f32, d_in[3]=gain (1,) f32. Output f32 (4,64,65536).
// Workspace layout (audio overlaps dead mixed buffer):
//   [0, SPEC)           spec   : 4*64*64*1025 cfloat = 134,348,800 B
//   [SPEC, SPEC+64Mi)   mixed  : 4*64*65536 f32      =  67,108,864 B (dead after fwd_fft)
//   [SPEC, SPEC+128Mi)  audio  : 4*64*64*2048 f32    = 134,217,728 B
// Total required ws: 268,566,528 B.
// ---------------------------------------------------------------------------
extern "C" void kernel_launch(void* const* d_in, const int* in_sizes, int n_in,
                              void* d_out, int out_size, void* d_ws, size_t ws_size,
                              hipStream_t stream) {
  const float* x        = (const float*)d_in[0];
  const float* transfer = (const float*)d_in[1];
  const float* mixer    = (const float*)d_in[2];
  const float* gain     = (const float*)d_in[3];
  float* out            = (float*)d_out;

  const size_t SPEC_BYTES = (size_t)BATCH * CPD * FRAMES * NCOEF * sizeof(float2);
  float2* spec = (float2*)d_ws;
  float*  mixed = (float*)((char*)d_ws + SPEC_BYTES);
  float*  audio = (float*)((char*)d_ws + SPEC_BYTES);  // overlaps mixed (dead), OK

  // K1: 8192 waves, each: 1 A-preload + 8 tiles x 16 chained f32 WMMAs
  mix_wmma<<<1024, 256, 0, stream>>>(x, mixer, mixed);
  // K2: one block per (b,d,frame) = 16384
  fwd_fft<<<BATCH * CPD * FRAMES, 256, 0, stream>>>(mixed, spec);
  // K3: 262400 threads = 1025 blocks
  scan_frames<<<(BATCH * CPD * NCOEF + 255) / 256, 256, 0, stream>>>(spec, transfer);
  // K4: one block per (b,d,frame)
  inv_fft<<<BATCH * CPD * FRAMES, 256, 0, stream>>>(spec, audio);
  // K5: one thread per output sample
  overlap_add<<<(int)(((size_t)BATCH * CPD * TIME + 255) / 256), 256, 0, stream>>>(
      audio, gain, out);
}